// GCNForGraph_47175920779584
// MI455X (gfx1250) — compile-verified
//
#include <hip/hip_runtime.h>
#include <hip/hip_bf16.h>

typedef float v2f __attribute__((ext_vector_type(2)));
typedef float v8f __attribute__((ext_vector_type(8)));

#define H_DIM 64
#define C_OUT 10
#define N_GRAPHS_C 512

// ---------------- utility ----------------
__global__ void k_zero(float* __restrict__ p, int n) {
    int i = blockIdx.x * blockDim.x + threadIdx.x;
    if (i < n) p[i] = 0.0f;
}

// ---------------- degree / norm ----------------
__global__ void k_degree(const int* __restrict__ dst, float* __restrict__ deg, int E) {
    int e = blockIdx.x * blockDim.x + threadIdx.x;
    if (e < E) atomicAdd(&deg[dst[e]], 1.0f);
}

__global__ void k_rsqrt(float* __restrict__ deg_dis, int n) {
    int i = blockIdx.x * blockDim.x + threadIdx.x;
    if (i < n) deg_dis[i] = rsqrtf(deg_dis[i] + 1.0f);  // deg+1 (self loop), dis = 1/sqrt
}

// ---------------- fp32 WMMA GEMM: H[M x 64] = A[M x K] @ W[K x 64] ----------------
// one wave (32 threads) per 16-row tile; 4 accumulators cover all 64 output cols.
// V_WMMA_F32_16X16X4_F32 fragment layouts per CDNA5 ISA 7.12.2:
//   A 16x4 : lanes 0-15 -> M=lane, VGPR0=K0, VGPR1=K1 ; lanes 16-31 -> K2,K3
//   B 4x16 : VGPR r: lanes 0-15 -> row K=r, lanes 16-31 -> row K=2+r ; col = lane&15
//   C/D 16x16: VGPR r: lanes 0-15 -> M=r, lanes 16-31 -> M=8+r ; col = lane&15
__global__ void k_gemm_wmma(const float* __restrict__ A, const float* __restrict__ W,
                            float* __restrict__ H, int M, int K) {
    const int row0 = blockIdx.x * 16;
    const int lane = threadIdx.x & 31;
    const int half = lane >> 4;      // 0 or 1
    const int l15  = lane & 15;

    v8f acc0 = {}, acc1 = {}, acc2 = {}, acc3 = {};

    int arow = row0 + l15;
    if (arow >= M) arow = M - 1;     // M is a multiple of 16 in practice; safe clamp
    const float* Abase = A + (size_t)arow * K;

    for (int k = 0; k < K; k += 4) {
        v2f a;
        a.x = Abase[k + 2 * half + 0];
        a.y = Abase[k + 2 * half + 1];

        const float* wrow = W + (size_t)(k + 2 * half) * H_DIM + l15;
        v2f b0, b1, b2, b3;
        b0.x = wrow[0];        b0.y = wrow[H_DIM + 0];
        b1.x = wrow[16];       b1.y = wrow[H_DIM + 16];
        b2.x = wrow[32];       b2.y = wrow[H_DIM + 32];
        b3.x = wrow[48];       b3.y = wrow[H_DIM + 48];

        acc0 = __builtin_amdgcn_wmma_f32_16x16x4_f32(false, a, false, b0, (short)0, acc0, false, false);
        acc1 = __builtin_amdgcn_wmma_f32_16x16x4_f32(false, a, false, b1, (short)0, acc1, false, false);
        acc2 = __builtin_amdgcn_wmma_f32_16x16x4_f32(false, a, false, b2, (short)0, acc2, false, false);
        acc3 = __builtin_amdgcn_wmma_f32_16x16x4_f32(false, a, false, b3, (short)0, acc3, false, false);
    }

#pragma unroll
    for (int r = 0; r < 8; ++r) {
        int m = row0 + r + 8 * half;
        if (m < M) {
            float* o = H + (size_t)m * H_DIM + l15;
            o[0]  = acc0[r];
            o[16] = acc1[r];
            o[32] = acc2[r];
            o[48] = acc3[r];
        }
    }
}

// ---------------- edge scatter: agg[dst] += h[src] * dis[src]*dis[dst] ----------------
// one thread handles 4 consecutive features of one edge (vectorized gather).
__global__ void k_scatter(const int* __restrict__ src, const int* __restrict__ dst,
                          const float* __restrict__ dis, const float* __restrict__ h,
                          float* __restrict__ agg, int E) {
    long long t = (long long)blockIdx.x * blockDim.x + threadIdx.x;
    if (t >= (long long)E * 16) return;
    int e = (int)(t >> 4);
    int f = (int)(t & 15) * 4;
    int s = src[e];
    int d = dst[e];
    float w = dis[s] * dis[d];
    const float4 v = *(const float4*)(h + (size_t)s * H_DIM + f);
    float* o = agg + (size_t)d * H_DIM + f;
    atomicAdd(o + 0, v.x * w);
    atomicAdd(o + 1, v.y * w);
    atomicAdd(o + 2, v.z * w);
    atomicAdd(o + 3, v.w * w);
}

// ---------------- combine self-loop + bias (+ReLU), in place into agg ----------------
__global__ void k_combine(float* __restrict__ agg, const float* __restrict__ h,
                          const float* __restrict__ dis, const float* __restrict__ b,
                          int relu, int Nn) {
    int t = blockIdx.x * blockDim.x + threadIdx.x;
    if (t >= Nn * H_DIM) return;
    int i = t >> 6;
    int f = t & 63;
    float sn = dis[i] * dis[i];
    float v = agg[t] + h[t] * sn + b[f];
    if (relu) v = fmaxf(v, 0.0f);
    agg[t] = v;
}

// ---------------- mean pool (atomic accumulate) ----------------
__global__ void k_pool(const float* __restrict__ feat, const int* __restrict__ batch,
                       float* __restrict__ sums, float* __restrict__ cnt, int Nn) {
    long long t = (long long)blockIdx.x * blockDim.x + threadIdx.x;
    if (t >= (long long)Nn * 16) return;
    int i = (int)(t >> 4);
    int f = (int)(t & 15) * 4;
    int g = batch[i];
    const float4 v = *(const float4*)(feat + (size_t)i * H_DIM + f);
    float* o = sums + (size_t)g * H_DIM + f;
    atomicAdd(o + 0, v.x);
    atomicAdd(o + 1, v.y);
    atomicAdd(o + 2, v.z);
    atomicAdd(o + 3, v.w);
    if ((t & 15) == 0) atomicAdd(&cnt[g], 1.0f);
}

// ---------------- final FC: out[g][c] = (sums[g]/cnt[g]) @ Wfc + bfc ----------------
__global__ void k_fc(const float* __restrict__ sums, const float* __restrict__ cnt,
                     const float* __restrict__ Wfc, const float* __restrict__ bfc,
                     float* __restrict__ out, int B) {
    int t = blockIdx.x * blockDim.x + threadIdx.x;
    if (t >= B * C_OUT) return;
    int g = t / C_OUT;
    int c = t % C_OUT;
    float inv = 1.0f / fmaxf(cnt[g], 1.0f);
    float acc = bfc[c];
#pragma unroll
    for (int f = 0; f < H_DIM; ++f)
        acc += sums[(size_t)g * H_DIM + f] * inv * Wfc[f * C_OUT + c];
    out[t] = acc;
}

// ---------------- host launcher ----------------
extern "C" void kernel_launch(void* const* d_in, const int* in_sizes, int n_in,
                              void* d_out, int out_size, void* d_ws, size_t ws_size,
                              hipStream_t stream) {
    (void)n_in; (void)ws_size; (void)out_size;

    const float* x   = (const float*)d_in[0];
    const int*   ei  = (const int*)d_in[1];     // [2, E] flat: src then dst
    const int*   bat = (const int*)d_in[2];
    const float* W1  = (const float*)d_in[3];
    const float* b1  = (const float*)d_in[4];
    const float* W2  = (const float*)d_in[5];
    const float* b2  = (const float*)d_in[6];
    const float* W3  = (const float*)d_in[7];
    const float* b3  = (const float*)d_in[8];
    const float* Wfc = (const float*)d_in[9];
    const float* bfc = (const float*)d_in[10];
    float* out = (float*)d_out;

    const int E = in_sizes[1] / 2;
    const int F = 128;
    const int N = in_sizes[0] / F;
    const int B = N_GRAPHS_C;

    const int* srcI = ei;
    const int* dstI = ei + E;

    // workspace layout (floats)
    float* ws  = (float*)d_ws;
    float* dis = ws;                              // N   (also holds deg before rsqrt)
    float* h   = dis + ((N + 255) & ~255);        // N*64
    float* agg = h + (size_t)N * H_DIM;           // N*64
    float* sums = agg + (size_t)N * H_DIM;        // B*64
    float* cnt  = sums + (size_t)B * H_DIM;       // B

    const int TB = 256;
    auto blk = [](long long n, int tb) { return (unsigned)((n + tb - 1) / tb); };

    // ---- init ----
    k_zero<<<blk(N, TB), TB, 0, stream>>>(dis, N);
    k_zero<<<blk((long long)N * H_DIM, TB), TB, 0, stream>>>(agg, N * H_DIM);
    k_zero<<<blk(B * H_DIM + B, TB), TB, 0, stream>>>(sums, B * H_DIM + B);

    // ---- degree + normalization ----
    k_degree<<<blk(E, TB), TB, 0, stream>>>(dstI, dis, E);
    k_rsqrt<<<blk(N, TB), TB, 0, stream>>>(dis, N);

    const long long scatterT = (long long)E * 16;
    const long long combT    = (long long)N * H_DIM;

    // ---- layer 1: h = x@W1 ; agg = scatter + self ; relu ----
    k_gemm_wmma<<<blk((long long)N, 16), 32, 0, stream>>>(x, W1, h, N, F);
    k_scatter<<<blk(scatterT, TB), TB, 0, stream>>>(srcI, dstI, dis, h, agg, E);
    k_combine<<<blk(combT, TB), TB, 0, stream>>>(agg, h, dis, b1, 1, N);

    // ---- layer 2 ----
    k_gemm_wmma<<<blk((long long)N, 16), 32, 0, stream>>>(agg, W2, h, N, H_DIM);
    k_zero<<<blk(combT, TB), TB, 0, stream>>>(agg, N * H_DIM);
    k_scatter<<<blk(scatterT, TB), TB, 0, stream>>>(srcI, dstI, dis, h, agg, E);
    k_combine<<<blk(combT, TB), TB, 0, stream>>>(agg, h, dis, b2, 1, N);

    // ---- layer 3 (no relu) ----
    k_gemm_wmma<<<blk((long long)N, 16), 32, 0, stream>>>(agg, W3, h, N, H_DIM);
    k_zero<<<blk(combT, TB), TB, 0, stream>>>(agg, N * H_DIM);
    k_scatter<<<blk(scatterT, TB), TB, 0, stream>>>(srcI, dstI, dis, h, agg, E);
    k_combine<<<blk(combT, TB), TB, 0, stream>>>(agg, h, dis, b3, 0, N);

    // ---- mean pool + FC ----
    k_pool<<<blk((long long)N * 16, TB), TB, 0, stream>>>(agg, bat, sums, cnt, N);
    k_fc<<<blk(B * C_OUT, TB), TB, 0, stream>>>(sums, cnt, Wfc, bfc, out, B);
}